// MultiSSM_13202729468281
// MI455X (gfx1250) — compile-verified
//
#include <hip/hip_runtime.h>
#include <math.h>

// Problem constants (from reference)
constexpr int kB    = 64;        // batch
constexpr int kNCP  = 512;       // n control planes
constexpr int kCPD  = 64;        // control plane dim
constexpr int kNF   = 256;       // n frames
constexpr int kWS   = 512;       // window size
constexpr int kSD   = 128;       // state dim
constexpr int kHOP  = kWS / 2;   // 256
constexpr int kNS   = kNF * kHOP;// 65536
constexpr int kTOPK = 8;

typedef float v2f __attribute__((ext_vector_type(2)));
typedef float v8f __attribute__((ext_vector_type(8)));

// CDNA5 f32 WMMA: D(16x16) = A(16x4) * B(4x16) + C, full f32 precision.
__device__ __forceinline__ v8f wmma4(v2f a, v2f b, v8f c) {
  return __builtin_amdgcn_wmma_f32_16x16x4_f32(
      /*neg_a=*/false, a, /*neg_b=*/false, b,
      /*c_mod=*/(short)0, c, /*reuse_a=*/false, /*reuse_b=*/false);
}

// ---------------------------------------------------------------------------
// K1: per-batch selection: argmax row of lookup, softmax stats, top-8,
//     times-argmax -> sample delay. One block per batch element.
// ---------------------------------------------------------------------------
__global__ __launch_bounds__(256) void k_select(
    const float* __restrict__ ctrl,      // (B, NCP)
    const float* __restrict__ times,     // (B, 1, NF)
    const float* __restrict__ items,     // (NCP, CPD*NF)
    int* __restrict__ imp_f, int* __restrict__ imp_c,
    float* __restrict__ imp_v,           // (B*TOPK)
    int* __restrict__ delay)             // (B)
{
  __shared__ float sv[256];
  __shared__ int   si[256];
  __shared__ int   selIdx[kTOPK];
  __shared__ float selVal[kTOPK];
  __shared__ float sMax, sSum;
  __shared__ int   sRow;
  const int b = blockIdx.x, t = threadIdx.x;
  const float NEG = -__builtin_inff();

  // argmax over control row (512) -> selects lookup row (one-hot @ items)
  {
    float v0 = ctrl[b * kNCP + t];
    float v1 = ctrl[b * kNCP + 256 + t];
    float mv; int mi;
    if (v1 > v0) { mv = v1; mi = 256 + t; } else { mv = v0; mi = t; }
    sv[t] = mv; si[t] = mi; __syncthreads();
    for (int s = 128; s; s >>= 1) {
      if (t < s && sv[t + s] > sv[t]) { sv[t] = sv[t + s]; si[t] = si[t + s]; }
      __syncthreads();
    }
    if (t == 0) sRow = si[0];
    __syncthreads();
  }
  const float* row = items + (size_t)sRow * (kCPD * kNF);

  // row max (softmax stability)
  {
    float mv = NEG;
    for (int i = t; i < kCPD * kNF; i += 256) mv = fmaxf(mv, row[i]);
    sv[t] = mv; __syncthreads();
    for (int s = 128; s; s >>= 1) { if (t < s) sv[t] = fmaxf(sv[t], sv[t + s]); __syncthreads(); }
    if (t == 0) sMax = sv[0];
    __syncthreads();
  }
  // sum exp
  {
    float acc = 0.f;
    for (int i = t; i < kCPD * kNF; i += 256) acc += __expf(row[i] - sMax);
    sv[t] = acc; __syncthreads();
    for (int s = 128; s; s >>= 1) { if (t < s) sv[t] += sv[t + s]; __syncthreads(); }
    if (t == 0) sSum = sv[0];
    __syncthreads();
  }
  // top-8 by 8 masked-argmax passes (softmax is monotone -> rank on raw values)
  for (int k = 0; k < kTOPK; k++) {
    float bv = NEG; int bi = -1;
    for (int i = t; i < kCPD * kNF; i += 256) {
      bool taken = false;
      #pragma unroll
      for (int q = 0; q < kTOPK; q++)
        if (q < k && selIdx[q] == i) taken = true;
      float x = row[i];
      if (!taken && x > bv) { bv = x; bi = i; }
    }
    sv[t] = bv; si[t] = bi; __syncthreads();
    for (int s = 128; s; s >>= 1) {
      if (t < s && sv[t + s] > sv[t]) { sv[t] = sv[t + s]; si[t] = si[t + s]; }
      __syncthreads();
    }
    if (t == 0) { selIdx[k] = si[0]; selVal[k] = sv[0]; }
    __syncthreads();
  }
  if (t < kTOPK) {
    int idx = selIdx[t];                      // flat = c*NF + f
    imp_c[b * kTOPK + t] = idx / kNF;
    imp_f[b * kTOPK + t] = idx % kNF;
    imp_v[b * kTOPK + t] = __expf(selVal[t] - sMax) / sSum;
  }
  // times argmax -> delta-conv == pure shift by argmax*(NS/NF)
  {
    float v = times[b * kNF + t];
    sv[t] = v; si[t] = t; __syncthreads();
    for (int s = 128; s; s >>= 1) {
      if (t < s && sv[t + s] > sv[t]) { sv[t] = sv[t + s]; si[t] = si[t + s]; }
      __syncthreads();
    }
    if (t == 0) delay[b] = si[0] * (kNS / kNF);
  }
}

// ---------------------------------------------------------------------------
// K2: generic WMMA GEMM  C(64 x N) = A(64 x K) @ W(K x N), f32.
//     Used for projB = proj@input_matrix and projD = proj@direct_matrix.
// ---------------------------------------------------------------------------
__global__ __launch_bounds__(128) void k_gemm64(
    const float* __restrict__ A, const float* __restrict__ W,
    float* __restrict__ C, int K, int N)
{
  const int lane = threadIdx.x & 31;
  const int wave = threadIdx.x >> 5;
  const int lo = lane & 15, hi = lane >> 4;
  const int m0 = wave * 16;
  const int n0 = blockIdx.x * 16;
  v8f c = {};
  for (int k = 0; k < K; k += 4) {
    v2f a, bv;
    a.x  = A[(m0 + lo) * K + k + 2 * hi];
    a.y  = A[(m0 + lo) * K + k + 2 * hi + 1];
    bv.x = W[(k + 2 * hi) * N + n0 + lo];
    bv.y = W[(k + 2 * hi + 1) * N + n0 + lo];
    c = wmma4(a, bv, c);
  }
  #pragma unroll
  for (int j = 0; j < 8; j++)
    C[(m0 + j + 8 * hi) * N + n0 + lo] = c[j];
}

// ---------------------------------------------------------------------------
// K3a: matrix squaring for transition powers: Pout(128x128) = Pin @ Pin.
//      Block = 8 waves (one 16-row tile each), grid.x = 8 (col tiles).
// ---------------------------------------------------------------------------
__global__ __launch_bounds__(256) void k_sq(
    const float* __restrict__ Pin, float* __restrict__ Pout)
{
  const int lane = threadIdx.x & 31;
  const int wave = threadIdx.x >> 5;
  const int lo = lane & 15, hi = lane >> 4;
  const int m0 = wave * 16;
  const int n0 = blockIdx.x * 16;
  v8f c = {};
  for (int k = 0; k < kSD; k += 4) {
    v2f a, bv;
    a.x  = Pin[(m0 + lo) * kSD + k + 2 * hi];
    a.y  = Pin[(m0 + lo) * kSD + k + 2 * hi + 1];
    bv.x = Pin[(k + 2 * hi) * kSD + n0 + lo];
    bv.y = Pin[(k + 2 * hi + 1) * kSD + n0 + lo];
    c = wmma4(a, bv, c);
  }
  #pragma unroll
  for (int j = 0; j < 8; j++)
    Pout[(m0 + j + 8 * hi) * kSD + n0 + lo] = c[j];
}

// ---------------------------------------------------------------------------
// K3b: scan base case T^(0)_f = XB_f = x_f @ Bm, assembled from the sparse
//      impulse table (<=8 rank-1 terms per batch) and precomputed projB.
// ---------------------------------------------------------------------------
__global__ __launch_bounds__(256) void k_xb(
    const float* __restrict__ projB,   // (CPD,SD)
    const int* __restrict__ imp_f, const int* __restrict__ imp_c,
    const float* __restrict__ imp_v,
    float* __restrict__ S0)            // (NF,B,SD)
{
  const int e = blockIdx.x * 256 + threadIdx.x;  // over NF*B*SD
  const int s  = e & (kSD - 1);
  const int bb = (e >> 7) & (kB - 1);
  const int f  = e >> 13;
  float acc = 0.f;
  #pragma unroll
  for (int q = 0; q < kTOPK; q++) {
    int ii = bb * kTOPK + q;
    if (imp_f[ii] == f) acc += imp_v[ii] * projB[imp_c[ii] * kSD + s];
  }
  S0[e] = acc;
}

// ---------------------------------------------------------------------------
// K3c: one Hillis-Steele doubling stage of the linear-recurrence scan:
//        f <  span: T'_f = T_f                 (already a full prefix)
//        f >= span: T'_f = T_f + T_{f-span} @ A^span
//      Grid (SD/16, NF), 4 waves per block (batch tiles). Fully parallel.
// ---------------------------------------------------------------------------
__global__ __launch_bounds__(128) void k_scan_stage(
    const float* __restrict__ Sin,     // (NF,B,SD)
    const float* __restrict__ Pw,      // A^span (SD,SD)
    float* __restrict__ Sout,          // (NF,B,SD)
    int span)
{
  const int f  = blockIdx.y;
  const int n0 = blockIdx.x * 16;
  const int lane = threadIdx.x & 31, wave = threadIdx.x >> 5;
  const int lo = lane & 15, hi = lane >> 4;
  const int m0 = wave * 16;

  const float* Tf = Sin + (size_t)f * kB * kSD;
  float* Of       = Sout + (size_t)f * kB * kSD;

  if (f < span) {                      // pass-through
    #pragma unroll
    for (int j = 0; j < 8; j++) {
      int rr = (m0 + j + 8 * hi) * kSD + n0 + lo;
      Of[rr] = Tf[rr];
    }
    return;
  }
  const float* Tp = Sin + (size_t)(f - span) * kB * kSD;
  __builtin_prefetch(&Tp[(m0 + lo) * kSD], 0, 3);   // global_prefetch_b8

  v8f c;
  #pragma unroll
  for (int j = 0; j < 8; j++)
    c[j] = Tf[(m0 + j + 8 * hi) * kSD + n0 + lo];

  for (int k = 0; k < kSD; k += 4) {
    v2f a, bv;
    a.x  = Tp[(m0 + lo) * kSD + k + 2 * hi];
    a.y  = Tp[(m0 + lo) * kSD + k + 2 * hi + 1];
    bv.x = Pw[(k + 2 * hi) * kSD + n0 + lo];
    bv.y = Pw[(k + 2 * hi + 1) * kSD + n0 + lo];
    c = wmma4(a, bv, c);
  }
  #pragma unroll
  for (int j = 0; j < 8; j++)
    Of[(m0 + j + 8 * hi) * kSD + n0 + lo] = c[j];
}

// ---------------------------------------------------------------------------
// K4: parallel over frames: out_f = S_f @ C + XD_f, Hann-windowed on store.
//     Grid (WS/16, NF), 4 waves per block (batch tiles).
// ---------------------------------------------------------------------------
__global__ __launch_bounds__(128) void k_frames(
    const float* __restrict__ states,  // (NF,B,SD)
    const float* __restrict__ Cmat,    // (SD,WS)
    const float* __restrict__ projD,   // (CPD,WS)
    const int* __restrict__ imp_f, const int* __restrict__ imp_c,
    const float* __restrict__ imp_v,
    float* __restrict__ frames)        // (NF,B,WS), windowed
{
  __shared__ int   lIF[kB * kTOPK];
  __shared__ int   lIC[kB * kTOPK];
  __shared__ float lIV[kB * kTOPK];
  const int t = threadIdx.x;
  for (int i = t; i < kB * kTOPK; i += 128) {
    lIF[i] = imp_f[i]; lIC[i] = imp_c[i]; lIV[i] = imp_v[i];
  }
  __syncthreads();

  const int f  = blockIdx.y;
  const int n0 = blockIdx.x * 16;
  const int lane = t & 31, wave = t >> 5;
  const int lo = lane & 15, hi = lane >> 4;
  const int m0 = wave * 16;
  const int col = n0 + lo;

  // C-init = sparse direct path x_f @ D via projD rows
  v8f c = {};
  #pragma unroll
  for (int j = 0; j < 8; j++) {
    int bb = m0 + j + 8 * hi;
    float acc = 0.f;
    #pragma unroll
    for (int q = 0; q < kTOPK; q++) {
      int ii = bb * kTOPK + q;
      if (lIF[ii] == f) acc += lIV[ii] * projD[lIC[ii] * kWS + col];
    }
    c[j] = acc;
  }
  const float* S = states + (size_t)f * kB * kSD;
  for (int k = 0; k < kSD; k += 4) {
    v2f a, bv;
    a.x  = S[(m0 + lo) * kSD + k + 2 * hi];
    a.y  = S[(m0 + lo) * kSD + k + 2 * hi + 1];
    bv.x = Cmat[(k + 2 * hi) * kWS + col];
    bv.y = Cmat[(k + 2 * hi + 1) * kWS + col];
    c = wmma4(a, bv, c);
  }
  const float win =
      0.5f * (1.0f - __cosf(6.28318530717958647692f * (float)col / (float)kWS));
  float* g = frames + (size_t)f * kB * kWS;
  #pragma unroll
  for (int j = 0; j < 8; j++)
    g[(m0 + j + 8 * hi) * kWS + col] = c[j] * win;
}

// ---------------------------------------------------------------------------
// K5: overlap-add (hop = WS/2 -> exactly two frames per sample) fused with
//     the Dirac-convolution shift. Fully writes d_out.
// ---------------------------------------------------------------------------
__global__ __launch_bounds__(256) void k_output(
    const float* __restrict__ frames,  // (NF,B,WS)
    const int* __restrict__ delay,     // (B)
    float* __restrict__ out)           // (B,1,NS)
{
  const int b = blockIdx.y;
  const int i = blockIdx.x * 256 + threadIdx.x;
  const int ts = i - delay[b];
  float v = 0.f;
  if (ts >= 0) {
    int f0 = ts >> 8;          // hop = 256
    int j0 = ts & 255;
    v = frames[((size_t)f0 * kB + b) * kWS + j0];
    if (f0 > 0)
      v += frames[((size_t)(f0 - 1) * kB + b) * kWS + j0 + 256];
  }
  out[(size_t)b * kNS + i] = v;
}

// ---------------------------------------------------------------------------
extern "C" void kernel_launch(void* const* d_in, const int* in_sizes, int n_in,
                              void* d_out, int out_size, void* d_ws, size_t ws_size,
                              hipStream_t stream) {
  const float* ctrl  = (const float*)d_in[0];  // (B, NCP)
  const float* times = (const float*)d_in[1];  // (B, 1, NF)
  const float* items = (const float*)d_in[2];  // (NCP, CPD*NF)
  const float* proj  = (const float*)d_in[3];  // (CPD, WS)
  const float* Amat  = (const float*)d_in[4];  // (SD, SD)
  const float* Bmat  = (const float*)d_in[5];  // (WS, SD)
  const float* Cmat  = (const float*)d_in[6];  // (SD, WS)
  const float* Dmat  = (const float*)d_in[7];  // (WS, WS)
  float* out = (float*)d_out;

  char* ws = (char*)d_ws;
  size_t off = 0;
  auto carve = [&](size_t bytes) -> void* {
    void* p = ws + off;
    off = (off + bytes + 255) & ~(size_t)255;
    return p;
  };
  int*   imp_f   = (int*)carve(kB * kTOPK * sizeof(int));
  int*   imp_c   = (int*)carve(kB * kTOPK * sizeof(int));
  float* imp_v   = (float*)carve(kB * kTOPK * sizeof(float));
  int*   delay   = (int*)carve(kB * sizeof(int));
  float* projB   = (float*)carve((size_t)kCPD * kSD * sizeof(float));    //  32 KB
  float* projD   = (float*)carve((size_t)kCPD * kWS * sizeof(float));    // 128 KB
  float* powers  = (float*)carve((size_t)7 * kSD * kSD * sizeof(float)); // 448 KB
  float* statesA = (float*)carve((size_t)kNF * kB * kSD * sizeof(float)); // 8 MB
  float* statesB = (float*)carve((size_t)kNF * kB * kSD * sizeof(float)); // 8 MB
  float* frames  = (float*)carve((size_t)kNF * kB * kWS * sizeof(float)); // 32 MB

  // 1) selection / softmax stats / top-8 / delay
  k_select<<<dim3(kB), dim3(256), 0, stream>>>(ctrl, times, items,
                                               imp_f, imp_c, imp_v, delay);
  // 2) projB = proj @ input_matrix ; projD = proj @ direct_matrix (WMMA)
  k_gemm64<<<dim3(kSD / 16), dim3(128), 0, stream>>>(proj, Bmat, projB, kWS, kSD);
  k_gemm64<<<dim3(kWS / 16), dim3(128), 0, stream>>>(proj, Dmat, projD, kWS, kWS);

  // 3) transition powers A^2 .. A^128 by repeated squaring (7 tiny GEMMs)
  {
    const float* pin = Amat;
    for (int j = 0; j < 7; j++) {
      float* pout = powers + (size_t)j * kSD * kSD;
      k_sq<<<dim3(kSD / 16), dim3(256), 0, stream>>>(pin, pout);
      pin = pout;
    }
  }
  // 4) scan base case T^(0) = XB (sparse impulses x projB)
  k_xb<<<dim3((kNF * kB * kSD) / 256), dim3(256), 0, stream>>>(
      projB, imp_f, imp_c, imp_v, statesA);

  // 5) 8 doubling stages: whole-chip parallel linear-recurrence scan.
  //    Ping-pong buffers; after 8 stages the full prefix lands in statesA.
  {
    float* bufs[2] = {statesA, statesB};
    for (int j = 0; j < 8; j++) {
      const float* Pw = (j == 0) ? Amat : powers + (size_t)(j - 1) * kSD * kSD;
      k_scan_stage<<<dim3(kSD / 16, kNF), dim3(128), 0, stream>>>(
          bufs[j & 1], Pw, bufs[(j + 1) & 1], 1 << j);
    }
  }
  // 6) parallel frame outputs + Hann window (WMMA)
  k_frames<<<dim3(kWS / 16, kNF), dim3(128), 0, stream>>>(statesA, Cmat, projD,
                                                          imp_f, imp_c, imp_v,
                                                          frames);
  // 7) overlap-add + impulse shift
  k_output<<<dim3(kNS / 256, kB), dim3(256), 0, stream>>>(frames, delay, out);
}